// PointerNetwork_6545530159661
// MI455X (gfx1250) — compile-verified
//
#include <hip/hip_runtime.h>
#include <math.h>

// Problem constants (match reference)
#define B_  32
#define S_  2048
#define D_  128
#define H_  512
#define FH  2048              // 4*H
#define KC  640               // D + H  (concat K for LSTM gemm)
#define NKQ_LSTM 160          // KC/4
#define NKQ_H    128          // H/4
#define NT_LSTM  128          // FH/16
#define NT_H     32           // H/16

#define NWG 16
#define TPB 256

typedef float v2f __attribute__((ext_vector_type(2)));
typedef float v8f __attribute__((ext_vector_type(8)));

// ---------------------------------------------------------------------------
// WMMA helper: D = A(16x4 f32) * B(4x16 f32) + C, full fp32 precision.
// ---------------------------------------------------------------------------
__device__ __forceinline__ v8f wmma4(v2f a, v2f b, v8f c) {
  return __builtin_amdgcn_wmma_f32_16x16x4_f32(false, a, false, b, (short)0, c,
                                               false, false);
}

// ---------------------------------------------------------------------------
// Grid-wide barrier over NWG persistent workgroups (L2 atomics, sense via gen)
// ---------------------------------------------------------------------------
__device__ __forceinline__ void grid_barrier(int* bar) {
  __syncthreads();
  if (threadIdx.x == 0) {
    int g0 = __hip_atomic_load(&bar[1], __ATOMIC_ACQUIRE, __HIP_MEMORY_SCOPE_AGENT);
    int v  = __hip_atomic_fetch_add(&bar[0], 1, __ATOMIC_ACQ_REL, __HIP_MEMORY_SCOPE_AGENT);
    if (v == NWG - 1) {
      __hip_atomic_store(&bar[0], 0, __ATOMIC_RELAXED, __HIP_MEMORY_SCOPE_AGENT);
      __hip_atomic_fetch_add(&bar[1], 1, __ATOMIC_ACQ_REL, __HIP_MEMORY_SCOPE_AGENT);
    } else {
      while (__hip_atomic_load(&bar[1], __ATOMIC_ACQUIRE, __HIP_MEMORY_SCOPE_AGENT) == g0) {
        __builtin_amdgcn_s_sleep(2);
      }
    }
  }
  __syncthreads();
}

__device__ __forceinline__ float sigmoidf_(float x) {
  return 1.0f / (1.0f + __expf(-x));
}

// ---------------------------------------------------------------------------
// Zero recurrent state + scores + barrier each call (ws is poisoned once).
// ---------------------------------------------------------------------------
__global__ void init_state(float* h, float* c, float* s, int* bar) {
  int i = blockIdx.x * TPB + threadIdx.x;
  if (i < 2 * B_ * H_) h[i] = 0.0f;
  if (i < B_ * H_)     c[i] = 0.0f;
  if (i < B_ * S_)     s[i] = 0.0f;
  if (i == 0) { bar[0] = 0; bar[1] = 0; }
}

// ---------------------------------------------------------------------------
// Pack row-major weights [W0 (k0rows x N); W1 ((K-k0rows) x N)] into the
// WMMA B-operand lane layout:  dst[(nt*nkq + kq)*32 + lane] = float2
//   n = nt*16 + (lane&15),  k = 4*kq + 2*(lane>>4)  -> {W[k][n], W[k+1][n]}
// ---------------------------------------------------------------------------
__global__ void pack_weights(const float* __restrict__ W0, int k0rows,
                             const float* __restrict__ W1, int N, int nkq,
                             float* __restrict__ dst, int total_f2) {
  int idx = blockIdx.x * TPB + threadIdx.x;
  if (idx >= total_f2) return;
  int lane = idx & 31;
  int kq   = (idx >> 5) % nkq;
  int nt   = (idx >> 5) / nkq;
  int n = nt * 16 + (lane & 15);
  int k = 4 * kq + ((lane >> 4) << 1);
  float a0 = (k     < k0rows) ? W0[(size_t)k * N + n]
                              : W1[(size_t)(k - k0rows) * N + n];
  float a1 = (k + 1 < k0rows) ? W0[(size_t)(k + 1) * N + n]
                              : W1[(size_t)(k + 1 - k0rows) * N + n];
  dst[2 * (size_t)idx]     = a0;
  dst[2 * (size_t)idx + 1] = a1;
}

// ---------------------------------------------------------------------------
// Persistent LSTM sequence kernel, slice-owned layout.
// 16 WGs; WG `wg` owns h-columns j in [wg*32, wg*32+32). Its 8 waves compute
// the 8 z-tiles feeding that slice: wave w -> gate (w>>1), subtile (w&1),
// global N-tile nt = gate*32 + wg*2 + (w&1).
// Per step:
//   stage A = [x_t | h] (80 KB) into LDS (cooperative, float4)
//   -> syncthreads -> WMMA GEMM (A from LDS, packed B from L2)
//   -> z tiles into LDS -> syncthreads -> gate math on the owned slice
//      (c lives in LDS for the whole sequence), h slice to global
//   -> ONE grid barrier.
// ---------------------------------------------------------------------------
__global__ void __launch_bounds__(TPB)
lstm_seq(const float* __restrict__ x, size_t xbstride, int T,
         const float* __restrict__ P, const float* __restrict__ bias,
         float* __restrict__ hbuf, float* __restrict__ cbuf,
         float* __restrict__ enc_out, int* bar) {
  __shared__ __align__(16) float A_lds[B_ * KC];        // 80 KB: [32][640]
  __shared__ __align__(16) float z_lds[4][B_][32];      // 16 KB
  __shared__ __align__(16) float c_lds[B_ * 32];        //  4 KB

  const int tid    = threadIdx.x;
  const int lane   = tid & 31;
  const int wave   = tid >> 5;
  const int wg     = blockIdx.x;                        // 0..15
  const int gate   = wave >> 1;                         // 0..3
  const int sub    = wave & 1;
  const int nt     = gate * 32 + wg * 2 + sub;          // global N-tile
  const int row_lo = lane & 15;
  const int ksub   = (lane >> 4) << 1;                  // K sub-offset 0 or 2
  const int rbase  = (lane >> 4) << 3;                  // C/D: M += 8 hi half
  const float* Pn  = P + (size_t)nt * NKQ_LSTM * 64;

  // Load owned c slice into LDS (persists across all T steps).
  #pragma unroll
  for (int k = 0; k < 4; ++k) {
    int u = tid + k * TPB;                // 0..1023
    int b_ = u >> 5, jj = u & 31;
    c_lds[u] = cbuf[(size_t)b_ * H_ + wg * 32 + jj];
  }

  for (int t = 0; t < T; ++t) {
    const float* hr = hbuf + (size_t)(t & 1) * (B_ * H_);
    const float* xt = x + (size_t)t * D_;

    // ---- Stage A = [x_t | h] into LDS (row-major [32][640]) ----
    for (int idx = tid; idx < (B_ * D_) / 4; idx += TPB) {       // x: 1024 f4
      int m = idx >> 5, kk = (idx & 31) << 2;
      *(float4*)&A_lds[m * KC + kk] = *(const float4*)(xt + (size_t)m * xbstride + kk);
    }
    for (int idx = tid; idx < (B_ * H_) / 4; idx += TPB) {       // h: 4096 f4
      int m = idx >> 7, kk = (idx & 127) << 2;
      *(float4*)&A_lds[m * KC + D_ + kk] = *(const float4*)(hr + (size_t)m * H_ + kk);
    }
    __syncthreads();

    // ---- WMMA GEMM: z-tile(nt) = A(32x640) @ P(nt)(640x16) ----
    v8f acc0 = {0.f, 0.f, 0.f, 0.f, 0.f, 0.f, 0.f, 0.f};
    v8f acc1 = {0.f, 0.f, 0.f, 0.f, 0.f, 0.f, 0.f, 0.f};
    for (int kq = 0; kq < NKQ_LSTM; ++kq) {
      int k0 = 4 * kq + ksub;
      v2f b  = *(const v2f*)(Pn + (size_t)kq * 64 + lane * 2);
      v2f a0 = *(const v2f*)&A_lds[row_lo * KC + k0];
      v2f a1 = *(const v2f*)&A_lds[(16 + row_lo) * KC + k0];
      acc0 = wmma4(a0, b, acc0);
      acc1 = wmma4(a1, b, acc1);
    }

    // Scatter z tile into LDS: C/D layout -> VGPR r: M=r / M=r+8 (hi lanes)
    int colL = sub * 16 + row_lo;         // 0..31 within slice
    #pragma unroll
    for (int r = 0; r < 8; ++r) {
      z_lds[gate][r + rbase][colL]      = acc0[r];
      z_lds[gate][16 + r + rbase][colL] = acc1[r];
    }
    __syncthreads();

    // ---- Gate math on owned slice (32 batches x 32 cols) ----
    float* hw = hbuf + (size_t)((t + 1) & 1) * (B_ * H_);
    #pragma unroll
    for (int k = 0; k < 4; ++k) {
      int u = tid + k * TPB;
      int b_ = u >> 5, jj = u & 31;
      int j = wg * 32 + jj;
      float zi = z_lds[0][b_][jj] + bias[j];
      float zf = z_lds[1][b_][jj] + bias[H_ + j];
      float zg = z_lds[2][b_][jj] + bias[2 * H_ + j];
      float zo = z_lds[3][b_][jj] + bias[3 * H_ + j];
      float cn = sigmoidf_(zf) * c_lds[u] + sigmoidf_(zi) * tanhf(zg);
      float hn = sigmoidf_(zo) * tanhf(cn);
      c_lds[u] = cn;
      hw[(size_t)b_ * H_ + j] = hn;
      if (enc_out)
        enc_out[(size_t)b_ * (S_ * H_) + (size_t)t * H_ + j] = hn;
    }
    __threadfence();
    grid_barrier(bar);   // single barrier per step: h visible to all WGs
  }

  // Persist owned c slice for the next launch (decoder continues from it).
  #pragma unroll
  for (int k = 0; k < 4; ++k) {
    int u = tid + k * TPB;
    int b_ = u >> 5, jj = u & 31;
    cbuf[(size_t)b_ * H_ + wg * 32 + jj] = c_lds[u];
  }
}

// ---------------------------------------------------------------------------
// dvec[b][n] = (h_d @ W_dec)[b][n] + b_dec[n] + b_enc[n]   (32 x 512, WMMA)
// ---------------------------------------------------------------------------
__global__ void dec_proj(const float* __restrict__ hd,
                         const float* __restrict__ Pdec,
                         const float* __restrict__ b_dec,
                         const float* __restrict__ b_enc,
                         float* __restrict__ dvec) {
  int lane  = threadIdx.x & 31;
  int gwave = blockIdx.x * (TPB / 32) + (threadIdx.x >> 5);  // 0..63
  int mt    = gwave >> 5;    // 0..1
  int ntile = gwave & 31;    // 0..31
  int row_lo = lane & 15, ksub = (lane >> 4) << 1, rbase = (lane >> 4) << 3;
  const float* Pn = Pdec + (size_t)ntile * NKQ_H * 64;
  const float* A  = hd + (size_t)mt * 16 * H_;
  v8f acc = {0.f, 0.f, 0.f, 0.f, 0.f, 0.f, 0.f, 0.f};
  for (int kq = 0; kq < NKQ_H; ++kq) {
    int k0 = 4 * kq + ksub;
    v2f b = *(const v2f*)(Pn + (size_t)kq * 64 + lane * 2);
    v2f a = *(const v2f*)(A + (size_t)row_lo * H_ + k0);
    acc = wmma4(a, b, acc);
  }
  int n = ntile * 16 + row_lo;
  float bb = b_dec[n] + b_enc[n];
  #pragma unroll
  for (int r = 0; r < 8; ++r) {
    int m = mt * 16 + r + rbase;
    dvec[(size_t)m * H_ + n] = acc[r] + bb;
  }
}

// ---------------------------------------------------------------------------
// scores[b][s] += sum_n tanh( (enc_out @ W_enc)[m][n] + dvec[b][n] ) * v_w[n]
// Block = one M-tile x 8 N-tiles; A panel (16x512 = 32 KB) staged in LDS and
// shared by the 8 waves (4x less L2 traffic on enc_out). Half-wave shfl_xor
// reduction over the 16 N-columns, one atomicAdd per row per VGPR.
// ---------------------------------------------------------------------------
__global__ void __launch_bounds__(TPB)
attn_scores(const float* __restrict__ enc_out, const float* __restrict__ Penc,
            const float* __restrict__ dvec, const float* __restrict__ vw,
            float* __restrict__ scores) {
  __shared__ __align__(16) float A_lds[16 * H_];   // 32 KB
  int lane  = threadIdx.x & 31;
  int wave  = threadIdx.x >> 5;
  int mt    = blockIdx.x >> 2;                     // 0..4095
  int ntile = (blockIdx.x & 3) * 8 + wave;         // 0..31
  int row_lo = lane & 15, ksub = (lane >> 4) << 1, rbase = (lane >> 4) << 3;
  const float* Pn = Penc + (size_t)ntile * NKQ_H * 64;
  const float* A  = enc_out + (size_t)mt * 16 * H_;

  for (int idx = threadIdx.x; idx < (16 * H_) / 4; idx += TPB)
    *(float4*)&A_lds[idx << 2] = *(const float4*)(A + ((size_t)idx << 2));
  __syncthreads();

  v8f acc = {0.f, 0.f, 0.f, 0.f, 0.f, 0.f, 0.f, 0.f};
  for (int kq = 0; kq < NKQ_H; ++kq) {
    int k0 = 4 * kq + ksub;
    v2f b = *(const v2f*)(Pn + (size_t)kq * 64 + lane * 2);
    v2f a = *(const v2f*)&A_lds[row_lo * H_ + k0];
    acc = wmma4(a, b, acc);
  }
  int n = ntile * 16 + row_lo;
  float v = vw[n];
  #pragma unroll
  for (int r = 0; r < 8; ++r) {
    int m  = mt * 16 + r + rbase;          // global row 0..65535
    int b_ = m >> 11, s = m & (S_ - 1);
    float val = tanhf(acc[r] + dvec[(size_t)b_ * H_ + n]) * v;
    val += __shfl_xor(val, 1, 32);
    val += __shfl_xor(val, 2, 32);
    val += __shfl_xor(val, 4, 32);
    val += __shfl_xor(val, 8, 32);
    if (row_lo == 0)
      atomicAdd(&scores[(size_t)b_ * S_ + s], val);
  }
}

// ---------------------------------------------------------------------------
// Row softmax over S=2048 (v_b is a constant shift -> cancels in softmax).
// ---------------------------------------------------------------------------
__global__ void softmax_rows(const float* __restrict__ scores,
                             float* __restrict__ out) {
  __shared__ float red[TPB];
  int b = blockIdx.x;
  const float* r = scores + (size_t)b * S_;
  float mx = -1e30f;
  for (int i = threadIdx.x; i < S_; i += TPB) mx = fmaxf(mx, r[i]);
  red[threadIdx.x] = mx; __syncthreads();
  for (int s = TPB / 2; s > 0; s >>= 1) {
    if (threadIdx.x < s) red[threadIdx.x] = fmaxf(red[threadIdx.x], red[threadIdx.x + s]);
    __syncthreads();
  }
  mx = red[0]; __syncthreads();
  float sum = 0.f;
  for (int i = threadIdx.x; i < S_; i += TPB) sum += __expf(r[i] - mx);
  red[threadIdx.x] = sum; __syncthreads();
  for (int s = TPB / 2; s > 0; s >>= 1) {
    if (threadIdx.x < s) red[threadIdx.x] += red[threadIdx.x + s];
    __syncthreads();
  }
  float inv = 1.f / red[0];
  for (int i = threadIdx.x; i < S_; i += TPB)
    out[(size_t)b * S_ + i] = __expf(r[i] - mx) * inv;
}

// ---------------------------------------------------------------------------
extern "C" void kernel_launch(void* const* d_in, const int* in_sizes, int n_in,
                              void* d_out, int out_size, void* d_ws,
                              size_t ws_size, hipStream_t stream) {
  (void)in_sizes; (void)n_in; (void)out_size; (void)ws_size;
  const float* x    = (const float*)d_in[0];   // (B,S,D)
  const float* st   = (const float*)d_in[1];   // (B,1,D)
  const float* Wxe  = (const float*)d_in[2];   // (D,4H)
  const float* Whe  = (const float*)d_in[3];   // (H,4H)
  const float* be   = (const float*)d_in[4];   // (4H,)
  const float* Wxd  = (const float*)d_in[5];
  const float* Whd  = (const float*)d_in[6];
  const float* bd   = (const float*)d_in[7];
  const float* Wenc = (const float*)d_in[8];   // (H,H)
  const float* benc = (const float*)d_in[9];
  const float* Wdec = (const float*)d_in[10];  // (H,H)
  const float* bdec = (const float*)d_in[11];
  const float* vw   = (const float*)d_in[12];  // (H,1)
  // d_in[13] = v_b : constant shift, cancels in softmax.

  float* ws = (float*)d_ws;
  size_t off = 0;
  float* P_e   = ws + off; off += (size_t)NT_LSTM * NKQ_LSTM * 64;  // 5.24 MB
  float* P_d   = ws + off; off += (size_t)NT_LSTM * NKQ_LSTM * 64;
  float* P_enc = ws + off; off += (size_t)NT_H * NKQ_H * 64;        // 1 MB
  float* P_dec = ws + off; off += (size_t)NT_H * NKQ_H * 64;
  float* hbuf  = ws + off; off += (size_t)2 * B_ * H_;              // ping-pong
  float* cbuf  = ws + off; off += (size_t)B_ * H_;
  float* dvec  = ws + off; off += (size_t)B_ * H_;
  float* scr   = ws + off; off += (size_t)B_ * S_;
  int*   bar   = (int*)(ws + off); off += 64;
  float* enc_out = ws + off; off += (size_t)B_ * S_ * H_;           // 128 MB

  init_state<<<(B_ * S_ + TPB - 1) / TPB, TPB, 0, stream>>>(hbuf, cbuf, scr, bar);

  pack_weights<<<(NT_LSTM * NKQ_LSTM * 32 + TPB - 1) / TPB, TPB, 0, stream>>>(
      Wxe, D_, Whe, FH, NKQ_LSTM, P_e, NT_LSTM * NKQ_LSTM * 32);
  pack_weights<<<(NT_LSTM * NKQ_LSTM * 32 + TPB - 1) / TPB, TPB, 0, stream>>>(
      Wxd, D_, Whd, FH, NKQ_LSTM, P_d, NT_LSTM * NKQ_LSTM * 32);
  pack_weights<<<(NT_H * NKQ_H * 32 + TPB - 1) / TPB, TPB, 0, stream>>>(
      Wenc, H_, Wenc, H_, NKQ_H, P_enc, NT_H * NKQ_H * 32);
  pack_weights<<<(NT_H * NKQ_H * 32 + TPB - 1) / TPB, TPB, 0, stream>>>(
      Wdec, H_, Wdec, H_, NKQ_H, P_dec, NT_H * NKQ_H * 32);

  // Encoder: 2048 steps; final h lands in hbuf[0] (even step count), c in cbuf.
  lstm_seq<<<NWG, TPB, 0, stream>>>(x, (size_t)S_ * D_, S_, P_e, be,
                                    hbuf, cbuf, enc_out, bar);
  // Decoder: 1 step from (h,c); h_d lands in hbuf[1].
  lstm_seq<<<NWG, TPB, 0, stream>>>(st, (size_t)D_, 1, P_d, bd,
                                    hbuf, cbuf, nullptr, bar);

  dec_proj<<<8, TPB, 0, stream>>>(hbuf + (size_t)B_ * H_, P_dec, bdec, benc, dvec);

  attn_scores<<<4096 * 4, TPB, 0, stream>>>(enc_out, P_enc, dvec, vw, scr);

  softmax_rows<<<B_, TPB, 0, stream>>>(scr, (float*)d_out);
}